// KNNCLayer_71966472011988
// MI455X (gfx1250) — compile-verified
//
#include <hip/hip_runtime.h>
#include <hip/hip_bf16.h>

// ---------------------------------------------------------------------------
// Fused KNN (largest-distance top-5) for MI455X / gfx1250.
//   Phase 1: tiled fp32 WMMA GEMM (V_WMMA_F32_16X16X4_F32) + streaming top-5,
//            with a double-buffered register-resident B tile so global loads
//            for tile t+1 overlap the WMMA + selection work of tile t.
//   Phase 2: merge partial top-5 lists, gather 5 neighbor rows, mean.
// ---------------------------------------------------------------------------

typedef __attribute__((ext_vector_type(2))) float v2f;
typedef __attribute__((ext_vector_type(8))) float v8f;

#define NQ       2048
#define NT       50000
#define NDIM     64
#define KNEIGH   5
#define TILE     16
#define KSTEPS   16          // 64 / 4
#define CHUNKS   25          // N split: 25 chunks x 2000 targets
#define TPC      2000        // targets per chunk
#define TILES_PC 125         // 2000 / 16
#define CANDS_PC 10          // 2 half-row lists x 5 per (row, chunk)

__global__ __launch_bounds__(32)
void knn_phase1(const float* __restrict__ X,   // [NQ][64]
                const float* __restrict__ T,   // [NT][64]
                float* __restrict__ ws_val,    // [NQ][CHUNKS][10]
                int*   __restrict__ ws_idx)    // [NQ][CHUNKS][10]
{
    const int lane  = threadIdx.x;      // 0..31
    const int l15   = lane & 15;
    const int half  = lane >> 4;        // 0 or 1
    const int mtile = blockIdx.x;       // 0..127
    const int chunk = blockIdx.y;       // 0..24

    __shared__ __align__(16) float lds_dist[TILE * TILE];

    // ---- Preload A tile (queries) in WMMA f32 16x16x4 A-layout -----------
    // Lane L holds X[m][4*s + 2*half + {0,1}] for each k-step s.
    const int    m    = mtile * TILE + l15;
    const float* xrow = X + (size_t)m * NDIM + 2 * half;
    v2f   a[KSTEPS];
    float x2part = 0.0f;
#pragma unroll
    for (int s = 0; s < KSTEPS; ++s) {
        a[s] = *(const v2f*)(xrow + 4 * s);
        x2part = fmaf(a[s].x, a[s].x, fmaf(a[s].y, a[s].y, x2part));
    }
    // Combine halves: lane L now holds |x_m|^2 for tile-row l15.
    const float x2 = x2part + __shfl_xor(x2part, 16, 32);
    // Broadcast the row norms this lane's accumulator rows need (r + 8*half).
    float x2row[8];
#pragma unroll
    for (int r = 0; r < 8; ++r)
        x2row[r] = __shfl(x2, r + 8 * half, 32);

    // ---- Streaming per-lane top-5 (descending) ---------------------------
    float bestv[KNEIGH];
    int   besti[KNEIGH];
#pragma unroll
    for (int j = 0; j < KNEIGH; ++j) { bestv[j] = -1.0f; besti[j] = 0; }

    const int n_base = chunk * TPC;

    // ---- Double-buffered register-resident B tile -------------------------
    v2f b[2][KSTEPS];
    {   // prologue: tile 0 loads (one clause of 16 x global_load_b64)
        const float* trow = T + (size_t)(n_base + l15) * NDIM + 2 * half;
#pragma unroll
        for (int s = 0; s < KSTEPS; ++s)
            b[0][s] = *(const v2f*)(trow + 4 * s);
    }

#pragma unroll 2
    for (int t = 0; t < TILES_PC; ++t) {
        const int cur = t & 1;
        const int nxt = cur ^ 1;

        // Issue tile t+1's loads NOW (overlap with WMMA + selection below).
        const int    tn     = (t + 1 < TILES_PC) ? (t + 1) : t;  // clamp tail
        const float* trow_n = T + (size_t)(n_base + tn * TILE + l15) * NDIM
                                + 2 * half;
        __builtin_prefetch(trow_n + TILE * NDIM, 0, 0);  // tile t+2 -> caches
#pragma unroll
        for (int s = 0; s < KSTEPS; ++s)
            b[nxt][s] = *(const v2f*)(trow_n + 4 * s);

        // ---- Compute current tile: 16 back-to-back fp32 WMMAs ------------
        const int n0 = n_base + t * TILE;
        v8f   acc    = {0.f, 0.f, 0.f, 0.f, 0.f, 0.f, 0.f, 0.f};
        float t2part = 0.0f;
#pragma unroll
        for (int s = 0; s < KSTEPS; ++s) {
            const v2f bb = b[cur][s];
            t2part = fmaf(bb.x, bb.x, fmaf(bb.y, bb.y, t2part));
            // D = A(16x4 f32) x B(4x16 f32) + C   (fp32 matrix pipe)
            acc = __builtin_amdgcn_wmma_f32_16x16x4_f32(
                      false, a[s], false, bb, (short)0, acc, false, false);
        }
        const float t2 = t2part + __shfl_xor(t2part, 16, 32); // |t_{n0+l15}|^2

        // dist tile -> LDS (transpose so each lane owns one row-half)
#pragma unroll
        for (int r = 0; r < 8; ++r) {
            const float d2   = x2row[r] + t2 - 2.0f * acc[r];
            const float dist = sqrtf(fmaxf(d2, 1e-12f));
            lds_dist[(r + 8 * half) * TILE + l15] = dist;
        }
        __syncthreads();

        // Lane covers row=l15, cols [8*half, 8*half+8)
        const float4 c0 = *(const float4*)&lds_dist[l15 * TILE + 8 * half];
        const float4 c1 = *(const float4*)&lds_dist[l15 * TILE + 8 * half + 4];
        const float cand[8] = {c0.x, c0.y, c0.z, c0.w, c1.x, c1.y, c1.z, c1.w};
#pragma unroll
        for (int c = 0; c < 8; ++c) {
            const float d = cand[c];
            if (d > bestv[KNEIGH - 1]) {
                bestv[KNEIGH - 1] = d;
                besti[KNEIGH - 1] = n0 + 8 * half + c;
#pragma unroll
                for (int j = KNEIGH - 2; j >= 0; --j) {
                    if (bestv[j + 1] > bestv[j]) {
                        float tv = bestv[j]; bestv[j] = bestv[j + 1]; bestv[j + 1] = tv;
                        int   ti = besti[j]; besti[j] = besti[j + 1]; besti[j + 1] = ti;
                    }
                }
            }
        }
        __syncthreads();
    }

    // ---- Emit partial lists: row l15, slot half*5 ------------------------
    const int    row  = mtile * TILE + l15;
    const size_t base = ((size_t)row * CHUNKS + chunk) * CANDS_PC + half * KNEIGH;
#pragma unroll
    for (int j = 0; j < KNEIGH; ++j) {
        ws_val[base + j] = bestv[j];
        ws_idx[base + j] = besti[j];
    }
}

__global__ __launch_bounds__(64)
void knn_phase2(const float* __restrict__ T,
                const float* __restrict__ ws_val,
                const int*   __restrict__ ws_idx,
                float* __restrict__ out)        // [NQ][64]
{
    const int row = blockIdx.x;
    __shared__ int sidx[KNEIGH];

    if (threadIdx.x == 0) {
        float bv[KNEIGH];
        int   bi[KNEIGH];
#pragma unroll
        for (int j = 0; j < KNEIGH; ++j) { bv[j] = -1.0f; bi[j] = 0x7fffffff; }
        const size_t base = (size_t)row * CHUNKS * CANDS_PC;
        for (int c = 0; c < CHUNKS * CANDS_PC; ++c) {
            const float d   = ws_val[base + c];
            const int   idx = ws_idx[base + c];
            if (d > bv[KNEIGH - 1] ||
                (d == bv[KNEIGH - 1] && idx < bi[KNEIGH - 1])) {
                bv[KNEIGH - 1] = d; bi[KNEIGH - 1] = idx;
                for (int j = KNEIGH - 2; j >= 0; --j) {
                    if (bv[j + 1] > bv[j] ||
                        (bv[j + 1] == bv[j] && bi[j + 1] < bi[j])) {
                        float tv = bv[j]; bv[j] = bv[j + 1]; bv[j + 1] = tv;
                        int   ti = bi[j]; bi[j] = bi[j + 1]; bi[j + 1] = ti;
                    }
                }
            }
        }
#pragma unroll
        for (int j = 0; j < KNEIGH; ++j) sidx[j] = bi[j];
    }
    __syncthreads();

    const int d = threadIdx.x;            // 0..63 = output dim
    float s = 0.0f;
#pragma unroll
    for (int j = 0; j < KNEIGH; ++j)
        s += T[(size_t)sidx[j] * NDIM + d];
    out[(size_t)row * NDIM + d] = s * (1.0f / KNEIGH);
}

extern "C" void kernel_launch(void* const* d_in, const int* in_sizes, int n_in,
                              void* d_out, int out_size, void* d_ws, size_t ws_size,
                              hipStream_t stream)
{
    (void)in_sizes; (void)n_in; (void)out_size; (void)ws_size;
    const float* X = (const float*)d_in[0];   // [2048, 64]
    const float* T = (const float*)d_in[1];   // [50000, 64]
    float* out = (float*)d_out;               // [2048, 64]

    // Workspace partition: 2 MB vals + 2 MB idx = 4 MB.
    const size_t ncand = (size_t)NQ * CHUNKS * CANDS_PC;
    float* ws_val = (float*)d_ws;
    int*   ws_idx = (int*)((char*)d_ws + ncand * sizeof(float));

    dim3 g1(NQ / TILE, CHUNKS);               // 128 x 25 waves
    knn_phase1<<<g1, 32, 0, stream>>>(X, T, ws_val, ws_idx);
    knn_phase2<<<NQ, 64, 0, stream>>>(T, ws_val, ws_idx, out);
}